// GraphSAGE_83872121356315
// MI455X (gfx1250) — compile-verified
//
#include <hip/hip_runtime.h>
#include <hip/hip_bf16.h>

typedef __attribute__((ext_vector_type(2))) float v2f;
typedef __attribute__((ext_vector_type(8))) float v8f;

// ---------------------------------------------------------------------------
// Utility kernels
// ---------------------------------------------------------------------------

__global__ void fill_zero_kernel(float* __restrict__ p, size_t n) {
    size_t i = (size_t)blockIdx.x * blockDim.x + threadIdx.x;
    size_t stride = (size_t)gridDim.x * blockDim.x;
    for (; i < n; i += stride) p[i] = 0.0f;
}

// one thread per edge: deg[dst] += 1
__global__ void degree_kernel(const int* __restrict__ ei, float* __restrict__ deg,
                              int nEdges) {
    int e = blockIdx.x * blockDim.x + threadIdx.x;
    if (e >= nEdges) return;
    int dst = ei[nEdges + e];
    unsafeAtomicAdd(&deg[dst], 1.0f);
}

// deg -> 1 / max(deg, 1)   (in place)
__global__ void invert_deg_kernel(float* __restrict__ deg, int nNodes) {
    int i = blockIdx.x * blockDim.x + threadIdx.x;
    if (i >= nNodes) return;
    deg[i] = 1.0f / fmaxf(deg[i], 1.0f);
}

// One wave32 per edge; lane g moves float4 chunk g of the 128-float feature row.
// agg[dst] += h[src]
__global__ void scatter_sum_kernel(const float* __restrict__ h,
                                   const int* __restrict__ ei,
                                   float* __restrict__ agg, int nEdges) {
    long long tid = (long long)blockIdx.x * blockDim.x + threadIdx.x;
    int e = (int)(tid >> 5);
    int g = (int)(tid & 31);
    if (e >= nEdges) return;
    int src = ei[e];
    int dst = ei[nEdges + e];
    const float4 v = *(const float4*)(h + (size_t)src * 128 + g * 4);
    float* p = agg + (size_t)dst * 128 + g * 4;
    unsafeAtomicAdd(p + 0, v.x);
    unsafeAtomicAdd(p + 1, v.y);
    unsafeAtomicAdd(p + 2, v.z);
    unsafeAtomicAdd(p + 3, v.w);
}

// ---------------------------------------------------------------------------
// Fused SAGEConv GEMM:
//   out[m, :] = relu?( (agg[m,:] * inv_deg[m]) @ Wl + h[m,:] @ Wr + bias )
// K = 256 logical (128 agg cols -> Wl rows, 128 h cols -> Wr rows), N = NT*16.
// Block = 256 threads = 8 waves; each wave computes 16 rows x N cols with
// V_WMMA_F32_16X16X4_F32. Weights staged in LDS in 64-row K-chunks with a
// padded stride (N+8) so the two K half-lanes hit disjoint bank ranges.
// ---------------------------------------------------------------------------
template <int NT, bool RELU>
__global__ __launch_bounds__(256) void sage_gemm_kernel(
    const float* __restrict__ h, const float* __restrict__ agg,
    const float* __restrict__ inv_deg,
    const float* __restrict__ Wl, const float* __restrict__ Wr,
    const float* __restrict__ bias,
    float* __restrict__ out, int nNodes) {
    constexpr int N   = NT * 16;
    constexpr int STR = N + 8;              // padded LDS row stride (floats)
    __shared__ float lds[64 * STR];

    const int tid  = threadIdx.x;
    const int lane = tid & 31;
    const int wave = tid >> 5;
    const int nl   = lane & 15;             // N index within tile / M row-in-wave
    const int kb   = (lane >> 4) * 2;       // K sub-offset: lanes 0-15 -> {0,1}, 16-31 -> {2,3}

    const int m0   = blockIdx.x * 128 + wave * 16;  // wave's first output row
    int rowA = m0 + nl;                      // A-fragment row for this lane
    if (rowA > nNodes - 1) rowA = nNodes - 1;  // clamp (keep EXEC all-ones for WMMA)
    const float invr = inv_deg[rowA];

    // Accumulators initialized with bias (C/D layout: col = nl for all 8 regs)
    v8f acc[NT];
#pragma unroll
    for (int nt = 0; nt < NT; ++nt) {
        const float b = bias[nt * 16 + nl];
        v8f t = {b, b, b, b, b, b, b, b};
        acc[nt] = t;
    }

    // 4 K-chunks of 64: chunks 0,1 -> (agg * inv_deg) x Wl ; chunks 2,3 -> h x Wr
#pragma unroll 1
    for (int chunk = 0; chunk < 4; ++chunk) {
        const float* __restrict__ W = (chunk < 2) ? Wl : Wr;
        const int krow = (chunk & 1) * 64;

        __syncthreads();
        // cooperative load of 64 x N weight slab into padded LDS
        for (int idx = tid; idx < 64 * N; idx += 256) {
            const int r = idx / N;
            const int c = idx - r * N;
            lds[r * STR + c] = W[(size_t)(krow + r) * N + c];
        }
        __syncthreads();

        const float* __restrict__ aptr =
            ((chunk < 2) ? agg : h) + (size_t)rowA * 128 + krow;
        const float scale = (chunk < 2) ? invr : 1.0f;

#pragma unroll
        for (int kk = 0; kk < 64; kk += 4) {
            // A fragment (16x4 fp32): lane holds A[row, kk+kb] and A[row, kk+kb+1]
            const float2 t = *(const float2*)(aptr + kk + kb);
            v2f a;
            a.x = t.x * scale;
            a.y = t.y * scale;
#pragma unroll
            for (int nt = 0; nt < NT; ++nt) {
                // B fragment (4x16 fp32): lane holds B[kk+kb, col], B[kk+kb+1, col]
                v2f b;
                b.x = lds[(kk + kb) * STR + nt * 16 + nl];
                b.y = lds[(kk + kb + 1) * STR + nt * 16 + nl];
                acc[nt] = __builtin_amdgcn_wmma_f32_16x16x4_f32(
                    /*neg_a=*/false, a, /*neg_b=*/false, b,
                    /*c_mod=*/(short)0, acc[nt],
                    /*reuse_a=*/false, /*reuse_b=*/false);
            }
        }
    }

    // Epilogue: D layout -> VGPR v holds M = m0 + v (+8 for lanes 16-31), col = nl
    const int mhalf = (lane >> 4) * 8;
#pragma unroll
    for (int nt = 0; nt < NT; ++nt) {
#pragma unroll
        for (int v = 0; v < 8; ++v) {
            const int m = m0 + mhalf + v;
            if (m < nNodes) {
                float val = acc[nt][v];
                if (RELU) val = fmaxf(val, 0.0f);
                out[(size_t)m * N + nt * 16 + nl] = val;
            }
        }
    }
}

// ---------------------------------------------------------------------------
// Launch: 3-layer GraphSAGE (mean aggregation), fp32 throughout.
// Inputs: 0:x 1:edge_index 2:Wl0 3:bl0 4:Wr0 5:Wl1 6:bl1 7:Wr1 8:Wl2 9:bl2 10:Wr2
// ---------------------------------------------------------------------------
extern "C" void kernel_launch(void* const* d_in, const int* in_sizes, int n_in,
                              void* d_out, int out_size, void* d_ws, size_t ws_size,
                              hipStream_t stream) {
    const float* x   = (const float*)d_in[0];
    const int*   ei  = (const int*)d_in[1];
    const float* Wl0 = (const float*)d_in[2];
    const float* bl0 = (const float*)d_in[3];
    const float* Wr0 = (const float*)d_in[4];
    const float* Wl1 = (const float*)d_in[5];
    const float* bl1 = (const float*)d_in[6];
    const float* Wr1 = (const float*)d_in[7];
    const float* Wl2 = (const float*)d_in[8];
    const float* bl2 = (const float*)d_in[9];
    const float* Wr2 = (const float*)d_in[10];
    float* out = (float*)d_out;

    const int nNodes = in_sizes[0] / 128;       // 50000
    const int nEdges = in_sizes[1] / 2;         // 800000
    const size_t featElems = (size_t)nNodes * 128;

    // Workspace layout (floats): [deg/inv_deg | agg | h1 | h2]
    float* ws  = (float*)d_ws;
    float* deg = ws;
    float* agg = ws + 51200;                    // padded past nNodes, 16B aligned
    float* h1  = agg + featElems;
    float* h2  = h1 + featElems;

    const int tpb = 256;
    const int zeroBlocks   = 2048;
    const int degBlocks    = (nNodes + tpb - 1) / tpb;
    const int edgeBlocks   = (nEdges + tpb - 1) / tpb;
    const int scatterBlocks = (int)(((long long)nEdges * 32 + tpb - 1) / tpb);
    const int gemmBlocks   = (nNodes + 127) / 128;

    // --- degrees (shared by all layers) ---
    fill_zero_kernel<<<degBlocks, tpb, 0, stream>>>(deg, (size_t)nNodes);
    degree_kernel<<<edgeBlocks, tpb, 0, stream>>>(ei, deg, nEdges);
    invert_deg_kernel<<<degBlocks, tpb, 0, stream>>>(deg, nNodes);

    // --- layer 0: x -> h1 (relu) ---
    fill_zero_kernel<<<zeroBlocks, tpb, 0, stream>>>(agg, featElems);
    scatter_sum_kernel<<<scatterBlocks, tpb, 0, stream>>>(x, ei, agg, nEdges);
    sage_gemm_kernel<8, true><<<gemmBlocks, tpb, 0, stream>>>(
        x, agg, deg, Wl0, Wr0, bl0, h1, nNodes);

    // --- layer 1: h1 -> h2 (relu) ---
    fill_zero_kernel<<<zeroBlocks, tpb, 0, stream>>>(agg, featElems);
    scatter_sum_kernel<<<scatterBlocks, tpb, 0, stream>>>(h1, ei, agg, nEdges);
    sage_gemm_kernel<8, true><<<gemmBlocks, tpb, 0, stream>>>(
        h1, agg, deg, Wl1, Wr1, bl1, h2, nNodes);

    // --- layer 2: h2 -> out (no relu, N = 64) ---
    fill_zero_kernel<<<zeroBlocks, tpb, 0, stream>>>(agg, featElems);
    scatter_sum_kernel<<<scatterBlocks, tpb, 0, stream>>>(h2, ei, agg, nEdges);
    sage_gemm_kernel<4, false><<<gemmBlocks, tpb, 0, stream>>>(
        h2, agg, deg, Wl2, Wr2, bl2, out, nNodes);
}